// EGCF_Encoder_35003983462570
// MI455X (gfx1250) — compile-verified
//
#include <hip/hip_runtime.h>
#include <stdint.h>

// EGCF encoder for MI455X (gfx1250, wave32).
// Memory-bound COO SpMM: async-to-LDS double-buffered row gather + L2 f32 atomics.
// Roofline: ~524 B/edge/SpMM at 128 FLOPs -> 0.24 FLOP/B; both dense matrices
// (25.6 MB + 12.8 MB) are L2-resident (192 MB), so gathers/atomics resolve in L2.

#define U_NUM 100000
#define I_NUM 50000
#define D_EMB 64
#define N_LAYERS 3
#define WAVES_PER_BLOCK 8

// ---------------- CDNA5 helpers ----------------

__device__ __forceinline__ float fast_tanh(float x) {
#if __has_builtin(__builtin_amdgcn_tanhf)
  return __builtin_amdgcn_tanhf(x);            // v_tanh_f32 (gfx1250 trans op)
#elif __has_builtin(__builtin_amdgcn_tanh_f32)
  return __builtin_amdgcn_tanh_f32(x);
#else
  float r;
  asm("v_tanh_f32 %0, %1" : "=v"(r) : "v"(x)); // hardware tanh; trans op
  return r;
#endif
}

// Async gather: 16 bytes per lane, global -> LDS, tracked by ASYNCcnt.
__device__ __forceinline__ void async_gather_b128(uint32_t lds_off, const void* gptr) {
  uint64_t ga = (uint64_t)(uintptr_t)gptr;
  asm volatile("global_load_async_to_lds_b128 %0, %1, off"
               :: "v"(lds_off), "v"(ga)
               : "memory");
}

__device__ __forceinline__ void wait_asynccnt_le1() {
#if __has_builtin(__builtin_amdgcn_s_wait_asynccnt)
  __builtin_amdgcn_s_wait_asynccnt(1);
#else
  asm volatile("s_wait_asynccnt 1" ::: "memory");
#endif
}

__device__ __forceinline__ void wait_asynccnt_le0() {
#if __has_builtin(__builtin_amdgcn_s_wait_asynccnt)
  __builtin_amdgcn_s_wait_asynccnt(0);
#else
  asm volatile("s_wait_asynccnt 0" ::: "memory");
#endif
}

// Low 32 bits of a flat shared-memory pointer == LDS byte offset (LDS aperture
// is {SHARED_BASE[63:32], offset[31:0]}), which is what the async VDST wants.
__device__ __forceinline__ uint32_t lds_offset(const void* p) {
  return (uint32_t)(uintptr_t)p;
}

// ---------------- kernels ----------------

// b128-vectorized zero (n need not be a multiple of 4; tail handled).
__global__ void zero_f32_kernel(float* __restrict__ p, long n) {
  const long n4 = n >> 2;
  float4* p4 = (float4*)p;
  long i = (long)blockIdx.x * blockDim.x + threadIdx.x;
  long s = (long)gridDim.x * blockDim.x;
  const float4 z = make_float4(0.f, 0.f, 0.f, 0.f);
  for (long j = i; j < n4; j += s) p4[j] = z;          // global_store_b128
  for (long j = (n4 << 2) + i; j < n; j += s) p[j] = 0.0f;
}

__global__ void degree_kernel(const int* __restrict__ rows, const int* __restrict__ cols,
                              float* __restrict__ dr, float* __restrict__ dc, int nE) {
  int i = blockIdx.x * blockDim.x + threadIdx.x;
  int s = gridDim.x * blockDim.x;
  for (; i < nE; i += s) {
    atomicAdd(&dr[rows[i]], 1.0f);                     // global_atomic_add_f32
    atomicAdd(&dc[cols[i]], 1.0f);
  }
}

__global__ void vals_kernel(const int* __restrict__ rows, const int* __restrict__ cols,
                            const float* __restrict__ dr, const float* __restrict__ dc,
                            float* __restrict__ vals, int nE) {
  int i = blockIdx.x * blockDim.x + threadIdx.x;
  int s = gridDim.x * blockDim.x;
  for (; i < nE; i += s) {
    float a = dr[rows[i]];
    float b = dc[cols[i]];
    float ia = (a > 0.0f) ? rsqrtf(a) : 0.0f;  // counts >=1 where >0, so max(.,1) is a no-op
    float ib = (b > 0.0f) ? rsqrtf(b) : 0.0f;
    vals[i] = ia * ib;
  }
}

// cur = tanh(cur); sum += cur   (b128-vectorized; n is a multiple of 4 here)
__global__ void tanh_acc_kernel(float* __restrict__ cur, float* __restrict__ sum, long n) {
  const long n4 = n >> 2;
  float4* c4 = (float4*)cur;
  float4* s4 = (float4*)sum;
  long i = (long)blockIdx.x * blockDim.x + threadIdx.x;
  long s = (long)gridDim.x * blockDim.x;
  for (long j = i; j < n4; j += s) {
    float4 c = c4[j];                                   // global_load_b128
    c.x = fast_tanh(c.x); c.y = fast_tanh(c.y);
    c.z = fast_tanh(c.z); c.w = fast_tanh(c.w);
    c4[j] = c;                                          // global_store_b128
    float4 a = s4[j];
    a.x += c.x; a.y += c.y; a.z += c.z; a.w += c.w;
    s4[j] = a;
  }
  for (long j = (n4 << 2) + i; j < n; j += s) {
    float t = fast_tanh(cur[j]);
    cur[j] = t;
    sum[j] += t;
  }
}

// y[dst[e]] += vals[e] * x[src[e]]  (rows are D_EMB=64 f32 = 256 B)
// One wave32 handles 2 edges per step: lanes 0-15 -> edge0, 16-31 -> edge1,
// each lane owns a 16 B chunk. Source rows are staged through LDS with
// global_load_async_to_lds_b128 (double-buffered, ASYNCcnt-tracked) so the
// gather for step k+1 overlaps the LDS read + atomic scatter of step k.
__global__ __launch_bounds__(256) void spmm_kernel(
    const int* __restrict__ dst_idx, const int* __restrict__ src_idx,
    const float* __restrict__ vals, const float* __restrict__ x,
    float* __restrict__ y, int nE) {
  __shared__ float4 stage[WAVES_PER_BLOCK * 64];  // per wave: 2 bufs x 32 float4 = 1 KB

  const int wave = threadIdx.x >> 5;
  const int lane = threadIdx.x & 31;
  const int half = lane >> 4;   // which edge of the pair
  const int q    = lane & 15;   // 16 B chunk within the 256 B row

  float4* slot0 = &stage[wave * 64 + lane];
  float4* slot1 = &stage[wave * 64 + 32 + lane];
  const uint32_t off0 = lds_offset(slot0);
  const uint32_t off1 = lds_offset(slot1);

  const int nPairs  = (nE + 1) >> 1;
  const int wstride = gridDim.x * WAVES_PER_BLOCK;
  int p = blockIdx.x * WAVES_PER_BLOCK + wave;
  if (p >= nPairs) return;      // wave-uniform: whole wave exits together

  // prologue: stage pair p into buffer 0
  {
    int e = p * 2 + half; if (e >= nE) e = nE - 1;
    int r = src_idx[e];
    async_gather_b128(off0, x + (long)r * D_EMB + q * 4);
  }

  int cur = 0;
  for (;;) {
    const int  pn   = p + wstride;
    const bool more = pn < nPairs;
    if (more) {  // stage next pair into the other buffer, then wait for current
      int e = pn * 2 + half; if (e >= nE) e = nE - 1;
      int r = src_idx[e];
      async_gather_b128(cur == 0 ? off1 : off0, x + (long)r * D_EMB + q * 4);
      wait_asynccnt_le1();
    } else {
      wait_asynccnt_le0();
    }

    const int  e  = p * 2 + half;
    const bool ok = e < nE;
    const int  ec = ok ? e : nE - 1;
    const float v = vals[ec];
    const int   d = dst_idx[ec];
    float4 t = (cur == 0) ? *slot0 : *slot1;   // ds_load_b128 from own slot

    if (ok) {
      float* yb = y + (long)d * D_EMB + q * 4;
      atomicAdd(yb + 0, v * t.x);              // global_atomic_add_f32 (L2-resident)
      atomicAdd(yb + 1, v * t.y);
      atomicAdd(yb + 2, v * t.z);
      atomicAdd(yb + 3, v * t.w);
    }

    if (!more) break;
    p = pn;
    cur ^= 1;
  }
}

// ---------------- host launcher ----------------

extern "C" void kernel_launch(void* const* d_in, const int* in_sizes, int n_in,
                              void* d_out, int out_size, void* d_ws, size_t ws_size,
                              hipStream_t stream) {
  const float* item_emb = (const float*)d_in[0];
  const int*   rows     = (const int*)d_in[1];
  const int*   cols     = (const int*)d_in[2];
  const int    nE       = in_sizes[1];

  const long UD = (long)U_NUM * D_EMB;   // 6.4M floats
  const long ID = (long)I_NUM * D_EMB;   // 3.2M floats

  // workspace layout (all 16B-aligned by construction): ~45.4 MB
  float* w    = (float*)d_ws;
  float* vals = w;  w += nE;
  float* dr   = w;  w += U_NUM;
  float* dc   = w;  w += I_NUM;
  float* ucur = w;  w += UD;
  float* icur = w;  w += ID;

  float* usum = (float*)d_out;       // [U, D]
  float* isum = usum + UD;           // [I, D]

  const int T = 256;
  zero_f32_kernel<<<256,  T, 0, stream>>>(dr, U_NUM);
  zero_f32_kernel<<<256,  T, 0, stream>>>(dc, I_NUM);
  zero_f32_kernel<<<2048, T, 0, stream>>>(usum, UD + ID);  // contiguous output

  degree_kernel<<<2048, T, 0, stream>>>(rows, cols, dr, dc, nE);
  vals_kernel<<<2048, T, 0, stream>>>(rows, cols, dr, dc, vals, nE);

  hipMemcpyAsync(icur, item_emb, ID * sizeof(float), hipMemcpyDeviceToDevice, stream);

  for (int l = 0; l < N_LAYERS; ++l) {
    // user = tanh(A @ item)
    zero_f32_kernel<<<2048, T, 0, stream>>>(ucur, UD);
    spmm_kernel<<<2048, T, 0, stream>>>(rows, cols, vals, icur, ucur, nE);
    tanh_acc_kernel<<<2048, T, 0, stream>>>(ucur, usum, UD);
    // item = tanh(A^T @ user)
    zero_f32_kernel<<<2048, T, 0, stream>>>(icur, ID);
    spmm_kernel<<<2048, T, 0, stream>>>(cols, rows, vals, ucur, icur, nE);
    tanh_acc_kernel<<<2048, T, 0, stream>>>(icur, isum, ID);
  }
}